// NeighborSearch_66606352827187
// MI455X (gfx1250) — compile-verified
//
#include <hip/hip_runtime.h>
#include <hip/hip_bf16.h>
#include <stdint.h>

typedef __attribute__((ext_vector_type(2))) float v2f;
typedef __attribute__((ext_vector_type(8))) float v8f;

#define N_DATA     32768
#define N_QUERIES  8192
#define N_DTILES   (N_DATA / 16)     // 2048 data tiles of 16
#define N_QTILES   (N_QUERIES / 16)  // 512 query tiles of 16
#define MAX_NEIGH  (N_QUERIES * 128) // 1048576

// ---- workspace layout (bytes) ----
static constexpr size_t OFF_BPACK = 0;                                                  // 2048*32 float2 = 512KB
static constexpr size_t OFF_APACK = OFF_BPACK + (size_t)N_DTILES * 32 * sizeof(float2); // 512*32 float2 = 128KB
static constexpr size_t OFF_THR   = OFF_APACK + (size_t)N_QTILES * 32 * sizeof(float2); // 8192 f32
static constexpr size_t OFF_CNT   = OFF_THR   + (size_t)N_QUERIES * sizeof(float);      // 8192 u32
static constexpr size_t OFF_OFFS  = OFF_CNT   + (size_t)N_QUERIES * sizeof(unsigned);   // 8192 u32

// ---------------------------------------------------------------------------
// Pack A (queries -> [qx,qy,qz,1]) and B (data -> [-2dx,-2dy,-2dz,|d|^2]) in
// the exact WMMA lane layout:
//   A 16x4 f32: VGPR0 = K0 (lanes 0-15) / K2 (lanes 16-31); VGPR1 = K1 / K3
//   B  4x16 f32: mirrored, N = lane%16
// Also compute per-query threshold thr = r^2 - |q|^2.
// ---------------------------------------------------------------------------
__global__ void ns_pack_kernel(const float* __restrict__ data,
                               const float* __restrict__ queries,
                               const float* __restrict__ radius,
                               float2* __restrict__ bpack,
                               float2* __restrict__ apack,
                               float*  __restrict__ thr) {
  int i = blockIdx.x * blockDim.x + threadIdx.x;
  if (i < N_DATA) {
    float x = data[3 * i + 0];
    float y = data[3 * i + 1];
    float z = data[3 * i + 2];
    int t = i >> 4, j = i & 15;
    bpack[t * 32 + j]      = make_float2(-2.0f * x, -2.0f * y);             // K0,K1
    bpack[t * 32 + 16 + j] = make_float2(-2.0f * z, x * x + y * y + z * z); // K2,K3
  }
  if (i < N_QUERIES) {
    float x = queries[3 * i + 0];
    float y = queries[3 * i + 1];
    float z = queries[3 * i + 2];
    int t = i >> 4, j = i & 15;
    apack[t * 32 + j]      = make_float2(x, y);     // K0,K1
    apack[t * 32 + 16 + j] = make_float2(z, 1.0f);  // K2,K3
    float r = radius[0];
    thr[i] = r * r - (x * x + y * y + z * z);
  }
}

// ---------------------------------------------------------------------------
// Pass 1: count neighbors per query. One wave32 owns one 16-query tile and
// sweeps all 2048 data tiles with V_WMMA_F32_16X16X4_F32.
// Counting is pure per-lane VALU (v_cmp + conditional add) so it co-issues
// with the matrix pipe; the cross-lane reduction (within each 16-lane half,
// since C/D rows split at lane 16) runs ONCE at loop exit as a 4-step
// shuffle-xor butterfly.
// ---------------------------------------------------------------------------
__global__ __launch_bounds__(256) void ns_count_kernel(
    const float2* __restrict__ bpack,
    const float2* __restrict__ apack,
    const float*  __restrict__ thr,
    unsigned*     __restrict__ counts) {
  const int lane  = threadIdx.x & 31;
  const int wtile = blockIdx.x * (blockDim.x >> 5) + (threadIdx.x >> 5); // [0,512)
  const int qbase = wtile * 16;

  float2 a2 = apack[wtile * 32 + lane];
  v2f a; a[0] = a2.x; a[1] = a2.y;

  // per-lane thresholds for the 8 rows this lane's accumulators touch
  float thrv[8];
  const int mbase = qbase + ((lane >> 4) << 3);
#pragma unroll
  for (int v = 0; v < 8; v++) thrv[v] = thr[mbase + v];

  unsigned cntv[8];
#pragma unroll
  for (int v = 0; v < 8; v++) cntv[v] = 0u;

  for (int t = 0; t < N_DTILES; t++) {
    float2 b2 = bpack[t * 32 + lane];
    v2f b; b[0] = b2.x; b[1] = b2.y;
    v8f c = {0.f, 0.f, 0.f, 0.f, 0.f, 0.f, 0.f, 0.f};
    c = __builtin_amdgcn_wmma_f32_16x16x4_f32(
        /*neg_a=*/false, a, /*neg_b=*/false, b,
        /*c_mod=*/(short)0, c, /*reuse_a=*/false, /*reuse_b=*/false);
#pragma unroll
    for (int v = 0; v < 8; v++) {
      cntv[v] += (c[v] <= thrv[v]) ? 1u : 0u; // per-lane, pure VALU
    }
  }

  // reduce across the 16 lanes of each half (xor distances 1,2,4,8 never
  // cross the lane-16 boundary). Lane 0 ends with rows 0-7; lane 16 rows 8-15.
#pragma unroll
  for (int v = 0; v < 8; v++) {
#pragma unroll
    for (int d = 1; d < 16; d <<= 1) {
      cntv[v] += (unsigned)__shfl_xor((int)cntv[v], d, 32);
    }
  }
  if (lane == 0) {
#pragma unroll
    for (int v = 0; v < 8; v++) counts[qbase + v] = cntv[v];
  }
  if (lane == 16) {
#pragma unroll
    for (int v = 0; v < 8; v++) counts[qbase + 8 + v] = cntv[v];
  }
}

// ---------------------------------------------------------------------------
// Exclusive scan of 8192 counts (single block, 1024 threads x 8 elems).
// Writes int64 row_splits[0..8192] and int32 emit offsets.
// ---------------------------------------------------------------------------
__global__ __launch_bounds__(1024) void ns_scan_kernel(
    const unsigned* __restrict__ counts,
    unsigned*       __restrict__ offsets,
    long long*      __restrict__ row_splits) {
  __shared__ unsigned sh[1024];
  const int tid = threadIdx.x;

  unsigned local[8];
  unsigned s = 0;
#pragma unroll
  for (int k = 0; k < 8; k++) {
    local[k] = counts[tid * 8 + k];
    s += local[k];
  }
  sh[tid] = s;
  __syncthreads();

  for (int off = 1; off < 1024; off <<= 1) {
    unsigned v = 0;
    if (tid >= off) v = sh[tid - off];
    __syncthreads();
    sh[tid] += v;
    __syncthreads();
  }

  unsigned run = (tid == 0) ? 0u : sh[tid - 1]; // exclusive prefix of this chunk
#pragma unroll
  for (int k = 0; k < 8; k++) {
    offsets[tid * 8 + k]    = run;
    row_splits[tid * 8 + k] = (long long)run;
    run += local[k];
  }
  if (tid == 1023) row_splits[N_QUERIES] = (long long)run;
}

// ---------------------------------------------------------------------------
// Fill padded neighbor slots with -1.
// ---------------------------------------------------------------------------
__global__ void ns_fill_kernel(long long* __restrict__ neigh) {
  int i = blockIdx.x * blockDim.x + threadIdx.x;
  if (i < MAX_NEIGH) neigh[i] = -1LL;
}

// ---------------------------------------------------------------------------
// Pass 2: emit CSR column indices. Same WMMA sweep; here the wave-uniform
// ballots ARE needed (rank = popc of lower same-row bits gives in-order
// compaction), and the s_bcnt/s_add pointer updates co-issue with VALU.
// ---------------------------------------------------------------------------
__global__ __launch_bounds__(256) void ns_emit_kernel(
    const float2*   __restrict__ bpack,
    const float2*   __restrict__ apack,
    const float*    __restrict__ thr,
    const unsigned* __restrict__ offsets,
    long long*      __restrict__ neigh) {
  const int lane  = threadIdx.x & 31;
  const int wtile = blockIdx.x * (blockDim.x >> 5) + (threadIdx.x >> 5);
  const int qbase = wtile * 16;

  float2 a2 = apack[wtile * 32 + lane];
  v2f a; a[0] = a2.x; a[1] = a2.y;

  float thrv[8];
  const int mbase = qbase + ((lane >> 4) << 3);
#pragma unroll
  for (int v = 0; v < 8; v++) thrv[v] = thr[mbase + v];

  unsigned p[16]; // per-row write pointers (wave-uniform values)
#pragma unroll
  for (int m = 0; m < 16; m++) p[m] = offsets[qbase + m];

  const unsigned lane_lo_mask = (lane < 16) ? ((1u << lane) - 1u) : 0u;
  const unsigned lane_hi_mask = (lane >= 16) ? ((1u << (lane - 16)) - 1u) : 0u;

  for (int t = 0; t < N_DTILES; t++) {
    float2 b2 = bpack[t * 32 + lane];
    v2f b; b[0] = b2.x; b[1] = b2.y;
    v8f c = {0.f, 0.f, 0.f, 0.f, 0.f, 0.f, 0.f, 0.f};
    c = __builtin_amdgcn_wmma_f32_16x16x4_f32(
        false, a, false, b, (short)0, c, false, false);
#pragma unroll
    for (int v = 0; v < 8; v++) {
      unsigned bal = __builtin_amdgcn_ballot_w32(c[v] <= thrv[v]); // uniform
      unsigned mlo = bal & 0xffffu;
      unsigned mhi = bal >> 16;
      if ((bal >> lane) & 1u) {
        unsigned pos;
        if (lane < 16) pos = p[v]     + (unsigned)__popc(mlo & lane_lo_mask);
        else           pos = p[v + 8] + (unsigned)__popc(mhi & lane_hi_mask);
        neigh[pos] = (long long)(t * 16 + (lane & 15));
      }
      p[v]     += (unsigned)__popc(mlo);
      p[v + 8] += (unsigned)__popc(mhi);
    }
  }
}

// ---------------------------------------------------------------------------
extern "C" void kernel_launch(void* const* d_in, const int* in_sizes, int n_in,
                              void* d_out, int out_size, void* d_ws, size_t ws_size,
                              hipStream_t stream) {
  const float* data    = (const float*)d_in[0];  // [32768,3]
  const float* queries = (const float*)d_in[1];  // [8192,3]
  const float* radius  = (const float*)d_in[2];  // scalar

  long long* neigh      = (long long*)d_out;              // [MAX_NEIGH]
  long long* row_splits = (long long*)d_out + MAX_NEIGH;  // [8193]

  char* ws = (char*)d_ws;
  float2*   bpack   = (float2*)(ws + OFF_BPACK);
  float2*   apack   = (float2*)(ws + OFF_APACK);
  float*    thr     = (float*)(ws + OFF_THR);
  unsigned* counts  = (unsigned*)(ws + OFF_CNT);
  unsigned* offsets = (unsigned*)(ws + OFF_OFFS);

  // 1) pack operands into WMMA lane layout (+ thresholds)
  ns_pack_kernel<<<(N_DATA + 255) / 256, 256, 0, stream>>>(
      data, queries, radius, bpack, apack, thr);

  // 2) count neighbors per query (WMMA sweep, VALU-only accumulation)
  ns_count_kernel<<<N_QTILES / 8, 256, 0, stream>>>(bpack, apack, thr, counts);

  // 3) scan -> row_splits (int64) + int32 offsets
  ns_scan_kernel<<<1, 1024, 0, stream>>>(counts, offsets, row_splits);

  // 4) pad neighbor slots with -1
  ns_fill_kernel<<<MAX_NEIGH / 256, 256, 0, stream>>>(neigh);

  // 5) emit CSR column indices (WMMA sweep + ballot-rank compaction)
  ns_emit_kernel<<<N_QTILES / 8, 256, 0, stream>>>(bpack, apack, thr, offsets, neigh);
}